// UNTRAINEDGCN_67250597921269
// MI455X (gfx1250) — compile-verified
//
#include <hip/hip_runtime.h>
#include <math.h>

// ---------------------------------------------------------------------------
// GCN pipeline for MI455X (gfx1250, wave32, WMMA + async global->LDS).
// ---------------------------------------------------------------------------

#define F_IN    128
#define HID     512
#define NGRAPHS 256

typedef __attribute__((ext_vector_type(2))) float v2f;
typedef __attribute__((ext_vector_type(8))) float v8f;

// ---------------------------------------------------------------------------
// Degree / normalization
// ---------------------------------------------------------------------------
__global__ void deg_init_kernel(float* __restrict__ deg, int n) {
    int i = blockIdx.x * blockDim.x + threadIdx.x;
    if (i < n) deg[i] = 1.0f;  // self-loop contributes 1
}

__global__ void deg_edges_kernel(const long long* __restrict__ dst,
                                 float* __restrict__ deg, int ne) {
    int e = blockIdx.x * blockDim.x + threadIdx.x;
    if (e < ne) atomicAdd(&deg[(int)dst[e]], 1.0f);
}

__global__ void make_dis_kernel(float* __restrict__ deg, int n) {
    int i = blockIdx.x * blockDim.x + threadIdx.x;
    if (i < n) deg[i] = rsqrtf(deg[i]);  // deg >= 1 always (self loops)
}

// ---------------------------------------------------------------------------
// GEMM: H[n, c] = sum_k X[n, k] * W[c, k]   (h = x @ W.T, no bias)
//
// Block = 256 threads = 8 waves; computes a 16-row x 128-col tile of H.
// The 16xK A-tile (shared by all 8 waves) is staged into LDS once with
// GLOBAL_LOAD_ASYNC_TO_LDS_B128 (ASYNCcnt), then each wave runs the K-loop
// with A-fragments from ds_load and B-fragments (its private 16 W rows,
// read exactly once, L2-resident) from global, feeding
// V_WMMA_F32_16X16X4_F32.
//
// A (16x4 f32): lanes 0-15 -> M=lane, K={k,k+1}; lanes 16-31 -> M=lane-16,
// K={k+2,k+3}. B (4x16): lane<16 -> K={k,k+1}, N=lane; lane>=16 ->
// K={k+2,k+3}, N=lane-16. C/D: VGPR v, lane<16 -> M=v; lane>=16 -> M=8+v.
// ---------------------------------------------------------------------------
template <int K>
__global__ __launch_bounds__(256) void gemm_wmma_kernel(
    const float* __restrict__ X, const float* __restrict__ W,
    float* __restrict__ H) {
    extern __shared__ float ldsA[];  // 16*K floats; dynamic LDS => base off 0

    const int tid     = threadIdx.x;
    const int wave    = tid >> 5;
    const int lane    = tid & 31;
    const int l15     = lane & 15;
    const int halfSel = lane >> 4;  // 0 or 1
    const int mTile   = blockIdx.x;
    const int nTile   = blockIdx.y * 8 + wave;

    // --- Stage A tile (16 rows x K cols, row-major) into LDS, async. ---
    {
        const unsigned long long gbase =
            (unsigned long long)(X + (long)mTile * 16 * K);
        const int totalBytes = 16 * K * 4;
        for (int off = tid * 16; off < totalBytes; off += 256 * 16) {
            unsigned int ldsOff = (unsigned int)off;  // byte offset in LDS
            unsigned int gOff   = (unsigned int)off;  // byte offset in global
            asm volatile("global_load_async_to_lds_b128 %0, %1, %2"
                         :
                         : "v"(ldsOff), "v"(gOff), "s"(gbase)
                         : "memory");
        }
        asm volatile("s_wait_asynccnt 0x0" ::: "memory");
    }
    __syncthreads();

    const float* arow = ldsA + l15 * K + 2 * halfSel;
    const float* wrow = W + (long)(nTile * 16 + l15) * K + 2 * halfSel;

    v8f c = {};
#pragma unroll 4
    for (int k = 0; k < K; k += 4) {
        v2f a = *(const v2f*)(arow + k);  // ds_load from staged tile
        v2f b = *(const v2f*)(wrow + k);  // global (read-once, L2 resident)
        __builtin_prefetch(wrow + k + 32, 0, 1);  // global_prefetch_b8
        c = __builtin_amdgcn_wmma_f32_16x16x4_f32(
            /*neg_a=*/false, a, /*neg_b=*/false, b,
            /*c_mod=*/(short)0, c, /*reuse_a=*/false, /*reuse_b=*/false);
    }

    float* outp = H + (long)(mTile * 16 + halfSel * 8) * HID + nTile * 16 + l15;
#pragma unroll
    for (int v = 0; v < 8; ++v) outp[(long)v * HID] = c[v];
}

// ---------------------------------------------------------------------------
// Scatter phase. Init: out[i] = b + dis[i]^2 * h[i]  (self-loop term).
// ---------------------------------------------------------------------------
__global__ __launch_bounds__(256) void conv_init_kernel(
    const float* __restrict__ Htmp, const float* __restrict__ bias,
    const float* __restrict__ dis, float* __restrict__ Out) {
    int i = blockIdx.x;
    float d2 = dis[i] * dis[i];
    for (int f = threadIdx.x; f < HID; f += 256)
        Out[(long)i * HID + f] = bias[f] + Htmp[(long)i * HID + f] * d2;
}

// Edge scatter: out[dst] += dis[src]*dis[dst] * h[src]. One block per edge,
// 128 threads x float4 = 512 features.
__global__ __launch_bounds__(128) void conv_edges_kernel(
    const float* __restrict__ Htmp, const long long* __restrict__ src,
    const long long* __restrict__ dst, const float* __restrict__ dis,
    float* __restrict__ Out) {
    int e = blockIdx.x;
    int s = (int)src[e];
    int d = (int)dst[e];
    float nrm = dis[s] * dis[d];
    int f = threadIdx.x * 4;
    float4 hv = *(const float4*)(Htmp + (long)s * HID + f);
    float* op = Out + (long)d * HID + f;
    atomicAdd(op + 0, hv.x * nrm);
    atomicAdd(op + 1, hv.y * nrm);
    atomicAdd(op + 2, hv.z * nrm);
    atomicAdd(op + 3, hv.w * nrm);
}

// ---------------------------------------------------------------------------
// Pooling: batch is sorted, so each graph g owns a contiguous node range
// found by binary search. out[g, 0:2048]   = segment_max(concat(h1..h4))
//          out[g, 2048:4096] = mean(h > 0)
// ---------------------------------------------------------------------------
__global__ __launch_bounds__(256) void pool_kernel(
    const float* __restrict__ h1, const float* __restrict__ h2,
    const float* __restrict__ h3, const float* __restrict__ h4,
    const long long* __restrict__ batch, float* __restrict__ out, int n) {
    int g = blockIdx.x;

    int lo = 0, hi = n;
    while (lo < hi) { int m = (lo + hi) >> 1; if (batch[m] < (long long)g) lo = m + 1; else hi = m; }
    int start = lo;
    hi = n;
    while (lo < hi) { int m = (lo + hi) >> 1; if (batch[m] < (long long)g + 1) lo = m + 1; else hi = m; }
    int end = lo;

    float invc = 1.0f / fmaxf((float)(end - start), 1.0f);
    const float* bufs[4] = {h1, h2, h3, h4};

    for (int f = threadIdx.x; f < 4 * HID; f += 256) {
        const float* hb = bufs[f >> 9] + (f & 511);
        float mx = -INFINITY;
        float pc = 0.0f;
        for (int i = start; i < end; ++i) {
            float v = hb[(long)i * HID];
            mx = fmaxf(mx, v);
            pc += (v > 0.0f) ? 1.0f : 0.0f;
        }
        out[(long)g * (8 * HID) + f]           = mx;
        out[(long)g * (8 * HID) + 4 * HID + f] = pc * invc;
    }
}

// ---------------------------------------------------------------------------
// Launch
// ---------------------------------------------------------------------------
extern "C" void kernel_launch(void* const* d_in, const int* in_sizes, int n_in,
                              void* d_out, int out_size, void* d_ws, size_t ws_size,
                              hipStream_t stream) {
    const float*     x     = (const float*)d_in[0];
    const long long* ei    = (const long long*)d_in[1];
    const long long* batch = (const long long*)d_in[2];
    // d_in[3]=lin_W, d_in[4]=lin_b: dead code in the reference (act=None).
    const float* W1 = (const float*)d_in[5];
    const float* b1 = (const float*)d_in[6];
    const float* W2 = (const float*)d_in[7];
    const float* b2 = (const float*)d_in[8];
    const float* W3 = (const float*)d_in[9];
    const float* b3 = (const float*)d_in[10];
    const float* W4 = (const float*)d_in[11];
    const float* b4 = (const float*)d_in[12];

    const int N = in_sizes[0] / F_IN;    // 20000 (divisible by 16)
    const int E = in_sizes[1] / 2;       // 320000

    const long long* srcp = ei;
    const long long* dstp = ei + E;

    // Workspace layout (floats): deg/dis[N] | tmp[N*HID] | h1..h4[N*HID]
    float* ws  = (float*)d_ws;
    float* deg = ws;
    float* tmp = deg + N;
    float* h1  = tmp + (long)N * HID;
    float* h2  = h1 + (long)N * HID;
    float* h3  = h2 + (long)N * HID;
    float* h4  = h3 + (long)N * HID;

    // Normalization dis = rsqrt(deg) stored in-place in deg.
    deg_init_kernel<<<(N + 255) / 256, 256, 0, stream>>>(deg, N);
    deg_edges_kernel<<<(E + 255) / 256, 256, 0, stream>>>(dstp, deg, E);
    make_dis_kernel<<<(N + 255) / 256, 256, 0, stream>>>(deg, N);

    const dim3 gemmGrid(N / 16, HID / 128);
    const size_t ldsA1 = 16 * F_IN * sizeof(float);  // 8 KB
    const size_t ldsAH = 16 * HID * sizeof(float);   // 32 KB

    // Layer 1 (K = F_IN = 128)
    gemm_wmma_kernel<F_IN><<<gemmGrid, 256, ldsA1, stream>>>(x, W1, tmp);
    conv_init_kernel<<<N, 256, 0, stream>>>(tmp, b1, deg, h1);
    conv_edges_kernel<<<E, 128, 0, stream>>>(tmp, srcp, dstp, deg, h1);

    // Layer 2 (K = HID)
    gemm_wmma_kernel<HID><<<gemmGrid, 256, ldsAH, stream>>>(h1, W2, tmp);
    conv_init_kernel<<<N, 256, 0, stream>>>(tmp, b2, deg, h2);
    conv_edges_kernel<<<E, 128, 0, stream>>>(tmp, srcp, dstp, deg, h2);

    // Layer 3
    gemm_wmma_kernel<HID><<<gemmGrid, 256, ldsAH, stream>>>(h2, W3, tmp);
    conv_init_kernel<<<N, 256, 0, stream>>>(tmp, b3, deg, h3);
    conv_edges_kernel<<<E, 128, 0, stream>>>(tmp, srcp, dstp, deg, h3);

    // Layer 4
    gemm_wmma_kernel<HID><<<gemmGrid, 256, ldsAH, stream>>>(h3, W4, tmp);
    conv_init_kernel<<<N, 256, 0, stream>>>(tmp, b4, deg, h4);
    conv_edges_kernel<<<E, 128, 0, stream>>>(tmp, srcp, dstp, deg, h4);

    // Pooling -> [NGRAPHS, 8*HID]
    pool_kernel<<<NGRAPHS, 256, 0, stream>>>(h1, h2, h3, h4, batch, (float*)d_out, N);
}